// CausalSelfAttention_84035330113750
// MI455X (gfx1250) — compile-verified
//
#include <hip/hip_runtime.h>
#include <stdint.h>
#include <stddef.h>

typedef __attribute__((ext_vector_type(2))) float v2f;
typedef __attribute__((ext_vector_type(8))) float v8f;
typedef __attribute__((ext_vector_type(4))) int   v4i;

#define B_   2
#define T_   2048
#define P_   2048
#define C_   1024
#define H_   16
#define HD_  64
#define S_   (P_ + T_)        // 4096
#define BT_  (B_ * T_)        // 4096
#define N3C_ (3 * C_)         // 3072

// ---- CDNA5 async global->LDS copy (guarded; falls back to sync copy) ------
#if __has_builtin(__builtin_amdgcn_global_load_async_to_lds_b128)
#define HAVE_ASYNC_LDS 1
#endif

__device__ __forceinline__ void async_cp16(void* lds, const void* g) {
#if defined(HAVE_ASYNC_LDS)
  __builtin_amdgcn_global_load_async_to_lds_b128(
      (v4i __attribute__((address_space(1)))*)(g),
      (v4i __attribute__((address_space(3)))*)(lds), 0, 0);
#else
  *(float4*)lds = *(const float4*)g;
#endif
}

#if __has_builtin(__builtin_amdgcn_s_wait_asynccnt)
#define ASYNC_WAIT(n) __builtin_amdgcn_s_wait_asynccnt(n)
#else
#define ASYNC_WAIT(n) asm volatile("s_wait_asynccnt %0" ::"n"(n) : "memory")
#endif

__device__ __forceinline__ v8f wmma4(v2f a, v2f b, v8f c) {
  // D = A(16x4 f32) * B(4x16 f32) + C(16x16 f32)
  return __builtin_amdgcn_wmma_f32_16x16x4_f32(false, a, false, b, (short)0, c,
                                               false, false);
}

// ---------------------------------------------------------------------------
// fp32 WMMA GEMM: C[M,N] = A[M,K] * B[K,N], N/K compile-time so all strides
// fold into immediate offsets. Block = 256 threads = 8 waves, tile 128x128.
// Wave w owns COLUMNS [w*16,(w+1)*16) x all 128 rows:
//   - B fragments: registers, prefetched one K-chunk ahead (coalesced global)
//   - A tile: double-buffered LDS (b64 conflict-free fragment reads),
//     register-prefetched one chunk ahead; one s_barrier per chunk
// ---------------------------------------------------------------------------
template <int N, int K>
__global__ __launch_bounds__(256) void wmma_gemm_f32(
    const float* __restrict__ A, const float* __restrict__ Bm,
    float* __restrict__ Cm) {
  __shared__ float As[2][128][20];  // pad 20: frag reads hit 16 distinct banks

  const int tid  = threadIdx.x;
  const int lane = tid & 31;
  const int w    = tid >> 5;      // wave id 0..7 -> column group
  const int nl   = lane & 15;
  const int hf   = lane >> 4;     // half: selects K pair within a 4-chunk

  const int m0 = blockIdx.y * 128;
  const int n0 = blockIdx.x * 128;

  const int ar = tid >> 2;        // A staging: row
  const int ac = (tid & 3) * 4;   // 16B column chunk

  v8f acc[8];
#pragma unroll
  for (int mt = 0; mt < 8; ++mt) acc[mt] = (v8f){0, 0, 0, 0, 0, 0, 0, 0};

  // running prefetch pointers (stride arithmetic is all compile-time)
  const float* aPtr = A + (size_t)(m0 + ar) * K + ac;
  const float* bPtr = Bm + n0 + w * 16 + nl;

  // prologue: prefetch chunk 0 (A -> regs for LDS, B -> operand regs)
  float4 aReg[2];
  v2f    bReg[4], bNxt[4];
#pragma unroll
  for (int i = 0; i < 2; ++i)
    aReg[i] = *(const float4*)(aPtr + (size_t)i * 64 * K);
#pragma unroll
  for (int kc = 0; kc < 4; ++kc) {
    bReg[kc].x = bPtr[(size_t)(kc * 4 + 2 * hf) * N];
    bReg[kc].y = bPtr[(size_t)(kc * 4 + 2 * hf + 1) * N];
  }

  int p = 0;
  for (int k0 = 0; k0 < K; k0 += 16) {
    // commit prefetched A chunk to LDS buffer p
#pragma unroll
    for (int i = 0; i < 2; ++i) *(float4*)&As[p][ar + i * 64][ac] = aReg[i];
    __syncthreads();

    // prefetch next chunk (latency hidden behind the 32-WMMA burst below)
    if (k0 + 16 < K) {
      aPtr += 16;
      bPtr += (size_t)16 * N;
#pragma unroll
      for (int i = 0; i < 2; ++i)
        aReg[i] = *(const float4*)(aPtr + (size_t)i * 64 * K);
#pragma unroll
      for (int kc = 0; kc < 4; ++kc) {
        bNxt[kc].x = bPtr[(size_t)(kc * 4 + 2 * hf) * N];
        bNxt[kc].y = bPtr[(size_t)(kc * 4 + 2 * hf + 1) * N];
      }
    }

#pragma unroll
    for (int kc = 0; kc < 4; ++kc) {
#pragma unroll
      for (int mt = 0; mt < 8; ++mt) {
        v2f a;  // A-frag: lane = row, half = K pair (contiguous b64 in LDS)
        a.x = As[p][mt * 16 + nl][kc * 4 + 2 * hf];
        a.y = As[p][mt * 16 + nl][kc * 4 + 2 * hf + 1];
        acc[mt] = wmma4(a, bReg[kc], acc[mt]);
      }
    }
#pragma unroll
    for (int kc = 0; kc < 4; ++kc) bReg[kc] = bNxt[kc];
    p ^= 1;  // next iteration stores into the other buffer; single barrier
  }

  // C/D layout: VGPR r -> row r + 8*half; lane -> column.
#pragma unroll
  for (int mt = 0; mt < 8; ++mt)
#pragma unroll
    for (int r = 0; r < 8; ++r)
      Cm[(size_t)(m0 + mt * 16 + r + 8 * hf) * N + n0 + w * 16 + nl] =
          acc[mt][r];
}

// ---------------------------------------------------------------------------
// Build k_full/v_full: [B,S,C] = concat(cache [B,P,C], qkv[:, C:2C / 2C:3C]).
// ---------------------------------------------------------------------------
__global__ __launch_bounds__(256) void kv_assemble(
    const float4* __restrict__ kc, const float4* __restrict__ vc,
    const float4* __restrict__ qkv, float4* __restrict__ kf,
    float4* __restrict__ vf) {
  const int C4 = C_ / 4;  // 256
  int i  = blockIdx.x * 256 + threadIdx.x;  // < B*S*C4 = 2^21
  int c4 = i & (C4 - 1);
  int s  = (i >> 8) & (S_ - 1);
  int b  = i >> 20;
  float4 kv, vv;
  if (s < P_) {
    size_t o = ((size_t)b * P_ + s) * C4 + c4;
    kv = kc[o];
    vv = vc[o];
  } else {
    size_t row = ((size_t)b * T_ + (s - P_)) * (N3C_ / 4);
    kv = qkv[row + (C_ / 4) + c4];
    vv = qkv[row + (2 * C_ / 4) + c4];
  }
  kf[i] = kv;
  vf[i] = vv;
}

// ---------------------------------------------------------------------------
// Flash attention fwd, fp32 WMMA. Block = 4 waves; one wave per 16 queries.
// Scores computed transposed (S^T = K_tile x Q^T): softmax (key) axis in
// VGPRs, query axis in lanes -> wave32 half-exchange reductions only.
// K tile: double-buffered per-wave LDS via async global->LDS copies.
// V tile: per-lane coalesced global prefetch into registers. Q: registers,
// doubling as the B-operand of S^T.
// ---------------------------------------------------------------------------
__device__ __forceinline__ void stage_k(float (*kd)[68], const float* kt,
                                        int lane) {
#pragma unroll
  for (int it = 0; it < 8; ++it) {
    int idx = it * 32 + lane;
    int row = idx >> 4, cc = (idx & 15) * 4;
    async_cp16(&kd[row][cc], kt + (size_t)row * C_ + cc);
  }
}

__global__ __launch_bounds__(128) void attn_fwd(
    const float* __restrict__ qkv,   // [BT, 3C]; Q = cols [h*64, h*64+64)
    const float* __restrict__ kf,    // [B, S, C]
    const float* __restrict__ vf,    // [B, S, C]
    float* __restrict__ yatt) {      // [BT, C]
  __shared__ float Ks[4][2][16][68];  // per-wave ping-pong K tiles (34 KB)

  const int tid  = threadIdx.x;
  const int lane = tid & 31;
  const int w    = tid >> 5;  // 0..3
  const int nl   = lane & 15;
  const int hf   = lane >> 4;

  const int qblk = blockIdx.x & 31;
  const int h    = (blockIdx.x >> 5) & 15;
  const int b    = blockIdx.x >> 9;
  const int qbase = qblk * 64 + w * 16;

  // Q fragment (B-operand of S^T = K x Q^T): lane = query, half = d pair
  const float* qp = qkv + ((size_t)(b * T_ + qbase + nl)) * N3C_ + h * HD_;
  v2f qf[16];
#pragma unroll
  for (int j = 0; j < 16; ++j) {
    qf[j].x = qp[4 * j + 2 * hf];
    qf[j].y = qp[4 * j + 2 * hf + 1];
  }

  v8f oacc[4];
#pragma unroll
  for (int dt = 0; dt < 4; ++dt) oacc[dt] = (v8f){0, 0, 0, 0, 0, 0, 0, 0};
  float mrun = -INFINITY, lrun = 0.0f;
  const float scale = 0.125f;  // 1/sqrt(64)

  const float* kbase = kf + (size_t)b * S_ * C_ + h * HD_;
  const float* vbase = vf + (size_t)b * S_ * C_ + h * HD_;
  const int limit = P_ + qbase + 16;  // exclusive causal key bound, 16-aligned

  stage_k(Ks[w][0], kbase, lane);  // tile 0 in flight
  int p = 0;
  for (int s0 = 0; s0 < limit; s0 += 16) {
    const bool more = (s0 + 16 < limit);  // wave-uniform
    if (more) {
      stage_k(Ks[w][p ^ 1], kbase + (size_t)(s0 + 16) * C_, lane);
      ASYNC_WAIT(8);  // 8 in flight for next tile; current tile complete
    } else {
      ASYNC_WAIT(0);
    }
    __builtin_amdgcn_wave_barrier();

    // ---- V tile prefetch into registers (coalesced; used after softmax)
    const float* vt = vbase + (size_t)s0 * C_;
    float vr[4][8];
#pragma unroll
    for (int kc = 0; kc < 4; ++kc)
#pragma unroll
      for (int dt = 0; dt < 4; ++dt) {
        vr[kc][2 * dt]     = vt[(size_t)(4 * kc + 2 * hf) * C_ + dt * 16 + nl];
        vr[kc][2 * dt + 1] =
            vt[(size_t)(4 * kc + 2 * hf + 1) * C_ + dt * 16 + nl];
      }

    // ---- S^T tile: rows = keys (VGPRs), cols = queries (lanes)
    float (*ksm)[68] = Ks[w][p];
    v8f st = (v8f){0, 0, 0, 0, 0, 0, 0, 0};
#pragma unroll
    for (int j = 0; j < 16; ++j) {
      v2f a;  // A-frag of K: lane = key row, half = d pair (contiguous b64)
      a.x = ksm[nl][4 * j + 2 * hf];
      a.y = ksm[nl][4 * j + 2 * hf + 1];
      st = wmma4(a, qf[j], st);
    }

    // ---- scale + causal mask (diagonal tile only)
    const bool last = (s0 + 16 >= limit);
    float sc[8];
#pragma unroll
    for (int r = 0; r < 8; ++r) {
      float v = st[r] * scale;  // key s0 + r + 8*hf, query qbase + nl
      if (last && (s0 + r + 8 * hf > P_ + qbase + nl)) v = -INFINITY;
      sc[r] = v;
    }

    // ---- online softmax per query column; merge halves via xor 16
    float tmax = sc[0];
#pragma unroll
    for (int r = 1; r < 8; ++r) tmax = fmaxf(tmax, sc[r]);
    tmax = fmaxf(tmax, __shfl_xor(tmax, 16, 32));
    float mnew  = fmaxf(mrun, tmax);
    float alpha = __expf(mrun - mnew);
    float p8[8], psum = 0.0f;
#pragma unroll
    for (int r = 0; r < 8; ++r) {
      p8[r] = __expf(sc[r] - mnew);
      psum += p8[r];
    }
    psum += __shfl_xor(psum, 16, 32);
    lrun = lrun * alpha + psum;
    mrun = mnew;

    // ---- rescale O rows: row q = r + 8*hf; alpha for q lives at lane q
#pragma unroll
    for (int r = 0; r < 8; ++r) {
      float aq = __shfl(alpha, r + 8 * hf, 32);
#pragma unroll
      for (int dt = 0; dt < 4; ++dt) oacc[dt][r] *= aq;
    }

    // ---- P A-fragments: lane needs all 16 key-probs of its query column
    float ps[16];
#pragma unroll
    for (int r = 0; r < 8; ++r) {
      float other = __shfl_xor(p8[r], 16, 32);
      ps[r]     = hf ? other : p8[r];  // compile-time indices (no spills)
      ps[r + 8] = hf ? p8[r] : other;
    }

    // ---- O += P x V_tile (V fragments from registers)
#pragma unroll
    for (int kc = 0; kc < 4; ++kc) {
      v2f a;  // A-frag of P: lane = query, half = key pair
      a.x = hf ? ps[4 * kc + 2] : ps[4 * kc];
      a.y = hf ? ps[4 * kc + 3] : ps[4 * kc + 1];
#pragma unroll
      for (int dt = 0; dt < 4; ++dt) {
        v2f bb;
        bb.x = vr[kc][2 * dt];
        bb.y = vr[kc][2 * dt + 1];
        oacc[dt] = wmma4(a, bb, oacc[dt]);
      }
    }
    __builtin_amdgcn_wave_barrier();
    p ^= 1;
  }

  // ---- normalize and store: element (dt, r, lane) -> row qbase + r + 8*hf
#pragma unroll
  for (int r = 0; r < 8; ++r) {
    float lq  = __shfl(lrun, r + 8 * hf, 32);
    float inv = 1.0f / lq;
    size_t row = (size_t)(b * T_ + qbase + r + 8 * hf);
#pragma unroll
    for (int dt = 0; dt < 4; ++dt)
      yatt[row * C_ + h * HD_ + dt * 16 + nl] = oacc[dt][r] * inv;
  }
}

// ---------------------------------------------------------------------------
extern "C" void kernel_launch(void* const* d_in, const int* in_sizes, int n_in,
                              void* d_out, int out_size, void* d_ws,
                              size_t ws_size, hipStream_t stream) {
  (void)in_sizes; (void)n_in; (void)out_size; (void)ws_size;
  const float* x       = (const float*)d_in[0];  // [B,T,C]
  const float* k_cache = (const float*)d_in[1];  // [B,P,C]
  const float* v_cache = (const float*)d_in[2];  // [B,P,C]
  const float* W_attn  = (const float*)d_in[3];  // [C,3C]
  const float* W_proj  = (const float*)d_in[4];  // [C,C]

  float* y  = (float*)d_out;                     // [B,T,C]
  float* kf = y + (size_t)BT_ * C_;              // [B,S,C]
  float* vf = kf + (size_t)B_ * S_ * C_;         // [B,S,C]

  float* qkv  = (float*)d_ws;                    // [BT, 3C] = 48 MB
  float* yatt = qkv + (size_t)BT_ * N3C_;        // [BT, C]  = 16 MB

  // 1) qkv = x @ W_attn
  wmma_gemm_f32<N3C_, C_><<<dim3(N3C_ / 128, BT_ / 128), 256, 0, stream>>>(
      x, W_attn, qkv);

  // 2) k_full / v_full = concat(cache, qkv k/v slices)
  kv_assemble<<<(B_ * S_ * (C_ / 4)) / 256, 256, 0, stream>>>(
      (const float4*)k_cache, (const float4*)v_cache, (const float4*)qkv,
      (float4*)kf, (float4*)vf);

  // 3) y_att = softmax(causal(Q K^T / sqrt(HD))) V
  attn_fwd<<<B_ * H_ * (T_ / 64), 128, 0, stream>>>(qkv, kf, vf, yatt);

  // 4) y = y_att @ W_proj
  wmma_gemm_f32<C_, C_><<<dim3(C_ / 128, BT_ / 128), 256, 0, stream>>>(
      yatt, W_proj, y);
}